// Sae_39410619908486
// MI455X (gfx1250) — compile-verified
//
#include <hip/hip_runtime.h>

// SAE forward (JumpReLU) for MI455X / gfx1250, wave32 + fp32 WMMA +
// async global->LDS double-buffered staging (ASYNCcnt pipeline).
//
//   pre    : xc = x - dec_b                         -> ws (4 MB)
//   encode : h = xc @ enc_w^T + enc_b               (M=512, N=65536, K=2048)
//            s = (h > exp(log_thr)), a = h * s      -> s to d_out, a to ws
//   decode : recon = (a @ dec_w^T + dec_b) / 64     (M=512, N=2048, K=65536)
//            loss  = sum((recon - x)^2) / (512*64)  -> fused, atomicAdd scalar
//
// fp32 WMMA (16x16x4) is chosen deliberately: the mask s compares h (sigma~8)
// against thr~0.064, so bf16's ~0.03 absolute error on h would flip mask bits.
// Compute-bound at ~300 TFLOP/s fp32-matrix; memory floor is only ~64 us.

#define D_IN   2048
#define D_SAE  65536
#define NSAMP  512

#define BM 64
#define BN 64
#define BK 32
#define LDSS 36   // 144B row stride: 16B-aligned rows for async B128 writes,
                  // and 36*r mod 64 distinct for r=0..15 -> conflict-free frags

#define NK_ENC (D_IN / BK)    // 64
#define NK_DEC (D_SAE / BK)   // 2048

typedef __attribute__((ext_vector_type(2))) float v2f;
typedef __attribute__((ext_vector_type(8))) float v8f;

// ---- CDNA5 async global->LDS copy (ASYNCcnt-tracked, no VGPR round trip) ----
__device__ __forceinline__ void async_ld16(uint32_t lds_byte_off,
                                           const float* gptr) {
  // GV mode: LDS byte offset from the first vgpr, 64-bit global address from VADDR
  asm volatile("global_load_async_to_lds_b128 %0, %1, off"
               :: "v"(lds_byte_off), "v"(gptr)
               : "memory");
}
__device__ __forceinline__ void wait_async_le8() {
  // 8 newest copies (next tile) may remain in flight; async loads complete
  // in order, so <=8 outstanding => previous tile's 8 copies have landed.
  asm volatile("s_wait_asynccnt 0x8" ::: "memory");
}
__device__ __forceinline__ void wait_async_0() {
  asm volatile("s_wait_asynccnt 0x0" ::: "memory");
}
__device__ __forceinline__ uint32_t lds_off(const void* p) {
  // generic AMDGPU LDS pointer = {aperture_hi32, lds_byte_offset}
  return (uint32_t)(uintptr_t)p;
}

// ------------------------------------------------------------ xc pre-pass ----
__global__ __launch_bounds__(256) void sae_xc_kernel(
    const float* __restrict__ x, const float* __restrict__ dec_b,
    float* __restrict__ xc) {
  const int i = blockIdx.x * 256 + threadIdx.x;       // float4 index
  const float4 xv = ((const float4*)x)[i];
  const float4 db = ((const float4*)dec_b)[i & (D_IN / 4 - 1)];
  float4 r;
  r.x = xv.x - db.x; r.y = xv.y - db.y;
  r.z = xv.z - db.z; r.w = xv.w - db.w;
  ((float4*)xc)[i] = r;
}

// ---------------------------------------------------------------- encode ----
__global__ __launch_bounds__(128) void sae_encode_kernel(
    const float* __restrict__ xc,      // [512, 2048]  (ws, pre-biased)
    const float* __restrict__ enc_w,   // [65536, 2048]
    const float* __restrict__ enc_b,   // [65536]
    const float* __restrict__ log_thr, // [65536]
    float* __restrict__ s_out,         // [512, 65536]
    float* __restrict__ a_out)         // [512, 65536] (ws)
{
  __shared__ __align__(16) float As[2][BM][LDSS];
  __shared__ __align__(16) float Bs[2][BN][LDSS];

  const int tid  = threadIdx.x;
  const int lane = tid & 31;
  const int wid  = tid >> 5;
  const int m0   = blockIdx.y * BM;
  const int n0   = blockIdx.x * BN;

  const int arow = (wid << 4) + (lane & 15);
  const int ncol = lane & 15;
  const int koff = (lane >> 4) << 1;

  // per-lane chunk mapping for async staging (4 b128 per lane per operand)
  v8f acc[4] = {};

  auto issue = [&](int k0, int buf) {
#pragma unroll
    for (int i = 0; i < 4; ++i) {
      const int c   = tid + i * 128;    // 0..511 float4 chunks
      const int row = c >> 3;
      const int col = (c & 7) << 2;
      async_ld16(lds_off(&As[buf][row][col]),
                 xc + (size_t)(m0 + row) * D_IN + k0 + col);
      async_ld16(lds_off(&Bs[buf][row][col]),
                 enc_w + (size_t)(n0 + row) * D_IN + k0 + col);
    }
  };

  issue(0, 0);
  for (int kc = 0; kc < NK_ENC; ++kc) {
    if (kc + 1 < NK_ENC) { issue((kc + 1) * BK, (kc + 1) & 1); wait_async_le8(); }
    else                 { wait_async_0(); }
    __syncthreads();                                  // tile kc visible to WG

    const float (*Ab)[LDSS] = As[kc & 1];
    const float (*Bb)[LDSS] = Bs[kc & 1];
#pragma unroll
    for (int kk = 0; kk < BK / 4; ++kk) {
      const int kb = (kk << 2) + koff;
      v2f af;
      af.x = Ab[arow][kb];
      af.y = Ab[arow][kb + 1];
#pragma unroll
      for (int t = 0; t < 4; ++t) {
        v2f bf;
        bf.x = Bb[(t << 4) + ncol][kb];
        bf.y = Bb[(t << 4) + ncol][kb + 1];
        acc[t] = __builtin_amdgcn_wmma_f32_16x16x4_f32(
            false, af, false, bf, (short)0, acc[t], false, false);
      }
    }
    __syncthreads();                    // all reads done before buf is reused
  }

#pragma unroll
  for (int t = 0; t < 4; ++t) {
    const int   n  = n0 + (t << 4) + ncol;
    const float eb = enc_b[n];
    const float th = expf(log_thr[n]);
#pragma unroll
    for (int v = 0; v < 8; ++v) {
      const int    m   = m0 + (wid << 4) + v + ((lane >> 4) << 3);
      const float  h   = acc[t][v] + eb;
      const float  sv  = (h > th) ? 1.0f : 0.0f;
      const size_t idx = (size_t)m * D_SAE + n;
      s_out[idx] = sv;
      a_out[idx] = h * sv;
    }
  }
}

// --------------------------------------------------- decode + fused loss ----
__global__ __launch_bounds__(128) void sae_decode_loss_kernel(
    const float* __restrict__ a,      // [512, 65536] (ws)
    const float* __restrict__ dec_w,  // [2048, 65536]
    const float* __restrict__ dec_b,  // [2048]
    const float* __restrict__ x,      // [512, 2048]
    float* __restrict__ loss)         // scalar (pre-zeroed)
{
  __shared__ __align__(16) float As[2][BM][LDSS];
  __shared__ __align__(16) float Bs[2][BN][LDSS];
  __shared__ float red[128];

  const int tid  = threadIdx.x;
  const int lane = tid & 31;
  const int wid  = tid >> 5;
  const int m0   = blockIdx.y * BM;
  const int n0   = blockIdx.x * BN;

  const int arow = (wid << 4) + (lane & 15);
  const int ncol = lane & 15;
  const int koff = (lane >> 4) << 1;

  v8f acc[4] = {};

  auto issue = [&](int k0, int buf) {
#pragma unroll
    for (int i = 0; i < 4; ++i) {
      const int c   = tid + i * 128;
      const int row = c >> 3;
      const int col = (c & 7) << 2;
      async_ld16(lds_off(&As[buf][row][col]),
                 a + (size_t)(m0 + row) * D_SAE + k0 + col);
      async_ld16(lds_off(&Bs[buf][row][col]),
                 dec_w + (size_t)(n0 + row) * D_SAE + k0 + col);
    }
  };

  issue(0, 0);
  for (int kc = 0; kc < NK_DEC; ++kc) {
    if (kc + 1 < NK_DEC) { issue((kc + 1) * BK, (kc + 1) & 1); wait_async_le8(); }
    else                 { wait_async_0(); }
    __syncthreads();

    const float (*Ab)[LDSS] = As[kc & 1];
    const float (*Bb)[LDSS] = Bs[kc & 1];
#pragma unroll
    for (int kk = 0; kk < BK / 4; ++kk) {
      const int kb = (kk << 2) + koff;
      v2f af;
      af.x = Ab[arow][kb];
      af.y = Ab[arow][kb + 1];
#pragma unroll
      for (int t = 0; t < 4; ++t) {
        v2f bf;
        bf.x = Bb[(t << 4) + ncol][kb];
        bf.y = Bb[(t << 4) + ncol][kb + 1];
        acc[t] = __builtin_amdgcn_wmma_f32_16x16x4_f32(
            false, af, false, bf, (short)0, acc[t], false, false);
      }
    }
    __syncthreads();
  }

  const float inv64 = 1.0f / 64.0f;
  float lsum = 0.0f;
#pragma unroll
  for (int t = 0; t < 4; ++t) {
    const int   n  = n0 + (t << 4) + ncol;
    const float db = dec_b[n];
#pragma unroll
    for (int v = 0; v < 8; ++v) {
      const int   m     = m0 + (wid << 4) + v + ((lane >> 4) << 3);
      const float recon = (acc[t][v] + db) * inv64;
      const float diff  = recon - x[(size_t)m * D_IN + n];
      lsum += diff * diff;
    }
  }

  red[tid] = lsum;
  __syncthreads();
  if (tid < 64) red[tid] += red[tid + 64];
  __syncthreads();
  if (tid < 32) red[tid] += red[tid + 32];
  __syncthreads();
  if (tid == 0) {
    float s = 0.0f;
    for (int i = 0; i < 32; ++i) s += red[i];
    atomicAdd(loss, s * (1.0f / (512.0f * 64.0f)));   // /N /X_MULT^2
  }
}

// -------------------------------------------------------------- launcher ----
extern "C" void kernel_launch(void* const* d_in, const int* in_sizes, int n_in,
                              void* d_out, int out_size, void* d_ws, size_t ws_size,
                              hipStream_t stream) {
  (void)in_sizes; (void)n_in; (void)ws_size;
  const float* x       = (const float*)d_in[0];
  const float* enc_w   = (const float*)d_in[1];
  const float* enc_b   = (const float*)d_in[2];
  const float* dec_w   = (const float*)d_in[3];
  const float* dec_b   = (const float*)d_in[4];
  const float* log_thr = (const float*)d_in[5];

  float* s_out = (float*)d_out;
  float* loss  = s_out + (size_t)(out_size - 1);
  float* a_ws  = (float*)d_ws;                        // 512*65536 floats
  float* xc_ws = a_ws + (size_t)NSAMP * D_SAE;        // 512*2048 floats

  hipMemsetAsync(loss, 0, sizeof(float), stream);

  sae_xc_kernel<<<(NSAMP * D_IN / 4) / 256, 256, 0, stream>>>(x, dec_b, xc_ws);

  sae_encode_kernel<<<dim3(D_SAE / BN, NSAMP / BM), 128, 0, stream>>>(
      xc_ws, enc_w, enc_b, log_thr, s_out, a_ws);

  sae_decode_loss_kernel<<<dim3(D_IN / BN, NSAMP / BM), 128, 0, stream>>>(
      a_ws, dec_w, dec_b, x, loss);
}